// points2mult_volume_55482387529812
// MI455X (gfx1250) — compile-verified
//
#include <hip/hip_runtime.h>
#include <stdint.h>

#define BOX    256
#define NCLS   2
#define NBATCH 2
#define NPTS   262144              // 2^18
#define VOL    (BOX * BOX * BOX)   // 16,777,216

// ---------------------------------------------------------------------------
// Hardware FP32 global atomic add, no return (STOREcnt path, L2 atomic units).
// Emits: global_atomic_add_f32 v[a:a+1], vdata, off scope:SCOPE_DEV
// ---------------------------------------------------------------------------
__device__ __forceinline__ void gatomic_add_f32(float* p, float v) {
    asm volatile("global_atomic_add_f32 %0, %1, off scope:SCOPE_DEV"
                 :: "v"(p), "v"(v)
                 : "memory");
}

// ---------------------------------------------------------------------------
// Zero-fill the 268 MB output via the CDNA5 async LDS->global store engine.
// Each of the 256 threads zeroes its own 16 B LDS slot with one ds_store_b128,
// then streams global_store_async_from_lds_b128 (ASYNCcnt) from that slot.
// Drained with s_wait_asynccnt before endpgm.
// ---------------------------------------------------------------------------
__global__ void __launch_bounds__(256) zero_async_kernel(float* __restrict__ out,
                                                         size_t nbytes) {
    __shared__ __align__(16) uint32_t zbuf[256 * 4];   // 4 KB of zeros

    // one b128 LDS store per thread: zbuf[tid*4 .. tid*4+3] = 0
    uint4 z4 = make_uint4(0u, 0u, 0u, 0u);
    *reinterpret_cast<uint4*>(&zbuf[threadIdx.x * 4]) = z4;
    __syncthreads();

    // Low 32 bits of a flat shared address == LDS byte offset on AMDGPU.
    uint32_t lds_off =
        (uint32_t)(uintptr_t)(&zbuf[0]) + (uint32_t)threadIdx.x * 16u;

    size_t tid    = (size_t)blockIdx.x * 256u + threadIdx.x;
    size_t stride = (size_t)gridDim.x * 256u;
    unsigned char* base = (unsigned char*)out;

    for (size_t off = tid * 16; off < nbytes; off += stride * 16) {
        unsigned char* gaddr = base + off;
        asm volatile("global_store_async_from_lds_b128 %0, %1, off"
                     :: "v"(gaddr), "v"(lds_off)
                     : "memory");
    }
    asm volatile("s_wait_asynccnt 0x0" ::: "memory");
}

// ---------------------------------------------------------------------------
// One thread per point: 8 trilinear corners x 2 classes of atomic adds.
// Out-of-range corners (reference sends them to idx 0 with weight 0) are
// simply skipped -- numerically identical, saves atomics.
// ---------------------------------------------------------------------------
__global__ void __launch_bounds__(256) cic_scatter_kernel(
        const float* __restrict__ points,
        const float* __restrict__ values,
        float* __restrict__ out) {
    uint32_t gid = blockIdx.x * 256u + threadIdx.x;   // < 524288
    uint32_t b = gid >> 18;              // gid / NPTS
    uint32_t n = gid & (NPTS - 1u);      // gid % NPTS

    // points[b][n][0..2] : 12 contiguous bytes (b64 + b32 loads)
    const float* pp = points + (size_t)gid * 3u;
    float px = (pp[0] + 0.5f) * (float)BOX;
    float py = (pp[1] + 0.5f) * (float)BOX;
    float pz = (pp[2] + 0.5f) * (float)BOX;

    float fbx = floorf(px), fby = floorf(py), fbz = floorf(pz);
    float rx = px - fbx, ry = py - fby, rz = pz - fbz;
    int ix = (int)fbx, iy = (int)fby, iz = (int)fbz;

    const float* vp = values + (size_t)b * (NCLS * NPTS) + n;
    float v0 = vp[0];        // class 0
    float v1 = vp[NPTS];     // class 1

    float wx[2] = {1.0f - rx, rx};
    float wy[2] = {1.0f - ry, ry};
    float wz[2] = {1.0f - rz, rz};
    int   cx[2] = {ix, ix + 1};
    int   cy[2] = {iy, iy + 1};
    int   cz[2] = {iz, iz + 1};
    bool  vx[2] = {(unsigned)cx[0] < BOX, (unsigned)cx[1] < BOX};
    bool  vy[2] = {(unsigned)cy[0] < BOX, (unsigned)cy[1] < BOX};
    bool  vz[2] = {(unsigned)cz[0] < BOX, (unsigned)cz[1] < BOX};

    float* out0 = out + (size_t)b * (NCLS * (size_t)VOL);  // class 0 volume
    float* out1 = out0 + VOL;                              // class 1 volume

#pragma unroll
    for (int dz = 0; dz < 2; ++dz) {
        if (!vz[dz]) continue;
        const uint32_t zoff = (uint32_t)cz[dz] * BOX;
#pragma unroll
        for (int dy = 0; dy < 2; ++dy) {
            if (!vy[dy]) continue;
            const uint32_t zyoff = (zoff + (uint32_t)cy[dy]) * BOX;
            const float wzy = wz[dz] * wy[dy];
#pragma unroll
            for (int dx = 0; dx < 2; ++dx) {
                if (!vx[dx]) continue;
                const float w = wzy * wx[dx];
                const uint32_t idx = zyoff + (uint32_t)cx[dx];
                gatomic_add_f32(out0 + idx, w * v0);
                gatomic_add_f32(out1 + idx, w * v1);
            }
        }
    }
}

// ---------------------------------------------------------------------------
extern "C" void kernel_launch(void* const* d_in, const int* in_sizes, int n_in,
                              void* d_out, int out_size, void* d_ws, size_t ws_size,
                              hipStream_t stream) {
    (void)in_sizes; (void)n_in; (void)d_ws; (void)ws_size;

    const float* points = (const float*)d_in[0];   // [B, N, 3]
    const float* values = (const float*)d_in[1];   // [B, C, N]
    float* out = (float*)d_out;                    // [B, C, box, box, box]

    const size_t nbytes = (size_t)out_size * sizeof(float);   // 268 MB

    // Pass 1: zero the output volume through the async store engine.
    zero_async_kernel<<<4096, 256, 0, stream>>>(out, nbytes);

    // Pass 2: trilinear scatter with hardware f32 atomics.
    const uint32_t total = NBATCH * NPTS;          // 524288
    cic_scatter_kernel<<<total / 256, 256, 0, stream>>>(points, values, out);
}